// DiagOUSDE_28003186769974
// MI455X (gfx1250) — compile-verified
//
#include <hip/hip_runtime.h>
#include <cstdint>
#include <cstddef>

// Problem constants (match reference: B=64, T=2048, D=256, fp32 everywhere).
#define BB 64
#define TT 2048
#define DD 256

// ---------------------------------------------------------------------------
// CDNA5 async global->LDS path (gfx1250). Prototype (probe-confirmed):
//   void __builtin_amdgcn_global_load_async_to_lds_b128(v4i*, v4i AS(3)*, Ii, Ii)
// ---------------------------------------------------------------------------
#if __has_builtin(__builtin_amdgcn_global_load_async_to_lds_b128) && \
    __has_builtin(__builtin_amdgcn_s_wait_asynccnt)
#define OU_USE_ASYNC 1
#else
#define OU_USE_ASYNC 0
#endif

typedef int v4i __attribute__((vector_size(4 * sizeof(int))));
typedef __attribute__((address_space(3))) v4i lds_v4i;

__device__ __forceinline__ void ou_copy16(const float* __restrict__ g, float* l) {
#if OU_USE_ASYNC
    // Generic pointer for the global source (numerically identical to AS1);
    // real addrspacecast for the LDS destination so the VGPR holds the LDS
    // byte offset.
    __builtin_amdgcn_global_load_async_to_lds_b128(
        (v4i*)g, (lds_v4i*)l, /*imm offset=*/0, /*cpol=*/0);
#else
    *(float4*)l = *(const float4*)g;
#endif
}

#if OU_USE_ASYNC
#define OU_WAIT_ASYNC(n) __builtin_amdgcn_s_wait_asynccnt(n)
#else
#define OU_WAIT_ASYNC(n) ((void)0)
#endif

// ---------------------------------------------------------------------------
// Kernel 2 (placed first so the disasm snippet shows it): the scan.
// grid=(B, D/DW), block=DW=64 (2 waves, wave32). Noise is triple-buffered
// through LDS via async DMA (prefetch distance 2); y carried in a register.
//
// Per-(b,d) recurrence, fully uniform because coefficient row 0 encodes the
// stationary init (A0=0, b0=mu, s0=sqrt(var0)):
//   y_t = A[t,d]*y_{t-1} + b[t,d] + s[t,d]*noise[b,t,d]
// ---------------------------------------------------------------------------
#define DW     64               // d-columns handled per block
#define CHUNK  64               // time rows staged per LDS buffer (16 KB)
#define NCH    (TT / CHUNK)     // 32 chunks, exact
#define NBUF   3                // triple buffering, prefetch distance 2
#define LD_IT  ((CHUNK * DW * 4) / (DW * 16))   // 16 b128 loads per thread/chunk

__global__ __launch_bounds__(DW)
void ou_scan_kernel(const float* __restrict__ noise,
                    const float* __restrict__ cA, const float* __restrict__ cB,
                    const float* __restrict__ cS, float* __restrict__ out) {
    __shared__ __align__(16) float sn[NBUF][CHUNK * DW];   // 3 x 16 KB

    const int b   = blockIdx.x;
    const int d0  = blockIdx.y * DW;
    const int tid = threadIdx.x;
    const int d   = d0 + tid;

    const float* nz = noise + (size_t)b * TT * DD + d0;   // row t at nz + t*DD
    float*       op = out   + (size_t)b * TT * DD + d0;

    // Issue one chunk's worth of async copies (rows [c*CHUNK, (c+1)*CHUNK)).
    auto issue_chunk = [&](int c) {
        const float* src = nz + (size_t)c * CHUNK * DD;
        float* dst = &sn[c % NBUF][0];
#pragma unroll
        for (int it = 0; it < LD_IT; ++it) {
            const int e   = it * DW + tid;   // 0..1023 16-byte units
            const int row = e >> 4;          // DW/4 = 16 units per row
            const int c16 = e & 15;
            ou_copy16(src + (size_t)row * DD + c16 * 4, dst + row * DW + c16 * 4);
        }
    };

    // Prologue: chunks 0 and 1 in flight (32 async instrs per wave).
    issue_chunk(0);
    issue_chunk(1);

    float y = 0.0f;   // A_0 = 0 makes the first step y = mu + s0*n0 regardless
    for (int c = 0; c < NCH; ++c) {
        const int buf = c % NBUF;
        // Keep two chunks ahead; wait immediates must be compile-time consts.
        if (c + 2 < NCH) {
            issue_chunk(c + 2);
            OU_WAIT_ASYNC(2 * LD_IT);   // chunk c done; c+1, c+2 in flight
        } else if (c + 1 < NCH) {
            OU_WAIT_ASYNC(LD_IT);       // chunk c done; c+1 in flight
        } else {
            OU_WAIT_ASYNC(0);           // drain
        }
        __syncthreads();                // other wave's rows are in LDS too

        const size_t base = (size_t)c * CHUNK * DD + d;
#pragma unroll 8
        for (int i = 0; i < CHUNK; ++i) {
            const size_t idx = base + (size_t)i * DD;
            const float A  = cA[idx];   // L2-resident (shared by all 64 b's)
            const float bb = cB[idx];
            const float s  = cS[idx];
            const float n  = sn[buf][i * DW + tid];
            y = fmaf(A, y, fmaf(s, n, bb));
            // Output is write-once, never re-read: non-temporal keeps the 192MB
            // L2 for the coefficient streams instead of the 134MB output.
            __builtin_nontemporal_store(y, &op[(size_t)(c * CHUNK + i) * DD + tid]);
        }
        __syncthreads();  // all reads of sn[buf] done before c+3 refills it
    }
}

// ---------------------------------------------------------------------------
// Kernel 1: per-(t,d) OU transition coefficients into workspace.
//   row 0   : A=0, b=mu[d], s=sqrt(sigma^2/(2 kappa))   (stationary init)
//   row t>0 : A=exp(-k dt), b=mu(1-A), s=sqrt(Q)
// ---------------------------------------------------------------------------
__device__ __forceinline__ float ou_softplus(float x) {
    return (x > 20.0f) ? x : log1pf(expf(x));
}

__global__ __launch_bounds__(256)
void ou_coeff_kernel(const float* __restrict__ ts, const float* __restrict__ mu,
                     const float* __restrict__ log_kappa, const float* __restrict__ log_sigma,
                     float* __restrict__ cA, float* __restrict__ cB, float* __restrict__ cS) {
    const int gid = blockIdx.x * blockDim.x + threadIdx.x;
    if (gid >= TT * DD) return;
    const int t = gid / DD;
    const int d = gid - t * DD;

    const float kap = ou_softplus(log_kappa[d]) + 1e-6f;
    const float sig = ou_softplus(log_sigma[d]) + 1e-6f;
    const float sig2 = sig * sig;

    float A, bb, s;
    if (t == 0) {
        A  = 0.0f;
        bb = mu[d];
        s  = sqrtf(sig2 / (2.0f * kap));          // sqrt(stationary variance)
    } else {
        const float dt = fmaxf(ts[t] - ts[t - 1], 1e-6f);
        A = expf(-kap * dt);
        const float tkd = 2.0f * kap * dt;
        const float Q_exact  = sig2 * (1.0f - expf(-tkd)) / fmaxf(2.0f * kap, 1e-12f);
        const float Q_taylor = sig2 * dt * (1.0f - kap * dt + tkd * tkd * (1.0f / 6.0f));
        const float Q = (tkd < 1e-6f) ? Q_taylor : Q_exact;
        bb = mu[d] * (1.0f - A);
        s  = sqrtf(Q);
    }
    cA[gid] = A;
    cB[gid] = bb;
    cS[gid] = s;
}

// ---------------------------------------------------------------------------
// Launcher. Inputs (setup_inputs order): ts[T], noise[B*T*D], mu[D],
// log_kappa[D], log_sigma[D] — all fp32. Output: [B,T,D] fp32.
// Workspace: 3 * T*D floats = 6 MB of coefficients.
// ---------------------------------------------------------------------------
extern "C" void kernel_launch(void* const* d_in, const int* in_sizes, int n_in,
                              void* d_out, int out_size, void* d_ws, size_t ws_size,
                              hipStream_t stream) {
    const float* ts        = (const float*)d_in[0];
    const float* noise     = (const float*)d_in[1];
    const float* mu        = (const float*)d_in[2];
    const float* log_kappa = (const float*)d_in[3];
    const float* log_sigma = (const float*)d_in[4];
    float* out = (float*)d_out;

    float* cA = (float*)d_ws;
    float* cB = cA + (size_t)TT * DD;
    float* cS = cB + (size_t)TT * DD;

    ou_coeff_kernel<<<(TT * DD) / 256, 256, 0, stream>>>(ts, mu, log_kappa, log_sigma,
                                                         cA, cB, cS);

    dim3 grid(BB, DD / DW);   // 64 x 4 = 256 blocks, 2 waves each
    ou_scan_kernel<<<grid, DW, 0, stream>>>(noise, cA, cB, cS, out);
}